// MPMLoss_51754355916968
// MI455X (gfx1250) — compile-verified
//
#include <hip/hip_runtime.h>

typedef float v2f __attribute__((ext_vector_type(2)));
typedef float v8f __attribute__((ext_vector_type(8)));

#define TILE 16
#define WAVES_PER_BLOCK 8

// For each query row i of Q [Nq,3], compute min_j ||q_i - r_j||^2 over R [Nr,3],
// clamped at 0.  outmin[b*Nq + i].
//
// WMMA embedding (V_WMMA_F32_16X16X4_F32, K=4):
//   A (16x4) row i  = [qx, qy, qz, 1]                 (register-resident per wave)
//   B (4x16) col j  = [-2rx, -2ry, -2rz, ||r||^2]     (staged in LDS once per block)
//   C (16x16)       = ||q_i||^2 broadcast along rows  (register-resident per wave)
//   D = A*B + C     = ||q||^2 - 2 q.r + ||r||^2       (full squared distance, no epilogue math)
//
// Each block: 8 waves = 8 query tiles (128 rows) of one batch; all waves share the
// LDS-cached transformed B tiles covering all Nr reference points.
// f32 operand layouts (wave32): A/B VGPR0 = K{0 | 2}, VGPR1 = K{1 | 3} split by
// lane-half; C/D VGPR v = row v (lanes 0-15) / row v+8 (lanes 16-31), col = lane&15.
__global__ void chamfer_min_kernel(const float* __restrict__ Q,
                                   const float* __restrict__ R,
                                   float* __restrict__ outmin,
                                   int Nq, int Nr) {
  extern __shared__ float ldsB[];  // [ (Nr/16) * 32 * 2 ] floats

  const int tid = threadIdx.x;
  const int lane = tid & 31;
  const int waveInBlk = tid >> 5;
  const int tilesPerBatch = Nq / TILE;
  const int blocksPerBatch = tilesPerBatch / WAVES_PER_BLOCK;
  const int b = blockIdx.x / blocksPerBatch;
  const int mtile = (blockIdx.x % blocksPerBatch) * WAVES_PER_BLOCK + waveInBlk;

  const float* __restrict__ Qb = Q + (size_t)b * Nq * 3;
  const float* __restrict__ Rb = R + (size_t)b * Nr * 3;
  const int ntiles = Nr / TILE;

  // ---- Stage transformed B operand for ALL reference tiles into LDS (once per block).
  // Entry (nt, l): lane l of tile nt; pair (b0, b1) = (K{0|2}, K{1|3}) per lane-half.
  const int totalEntries = ntiles * 32;
  for (int idx = tid; idx < totalEntries; idx += (WAVES_PER_BLOCK * 32)) {
    const int l = idx & 31;
    const int nt = idx >> 5;
    const bool ehi = (l >= 16);
    const int col = nt * TILE + (l & 15);
    const float gx = Rb[col * 3 + 0];
    const float gy = Rb[col * 3 + 1];
    const float gz = Rb[col * 3 + 2];
    const float n2 = gx * gx + gy * gy + gz * gz;
    ldsB[idx * 2 + 0] = ehi ? (-2.0f * gz) : (-2.0f * gx);
    ldsB[idx * 2 + 1] = ehi ? n2 : (-2.0f * gy);
  }

  // ---- A matrix (overlaps LDS-fill latency, before the barrier)
  const bool hi = (lane >= 16);
  const int qrow = mtile * TILE + (lane & 15);
  const float qx = Qb[qrow * 3 + 0];
  const float qy = Qb[qrow * 3 + 1];
  const float qz = Qb[qrow * 3 + 2];
  v2f a;
  a.x = hi ? qz : qx;
  a.y = hi ? 1.0f : qy;

  // ---- C base: VGPR v holds ||q_{mtile*16 + v + (hi?8:0)}||^2
  v8f cbase;
#pragma unroll
  for (int v = 0; v < 8; ++v) {
    const int r = mtile * TILE + v + (hi ? 8 : 0);
    const float x = Qb[r * 3 + 0];
    const float y = Qb[r * 3 + 1];
    const float z = Qb[r * 3 + 2];
    cbase[v] = x * x + y * y + z * z;
  }

  __syncthreads();

  // ---- Main sweep: ds_load_b64 + v_wmma + fmin per tile; two accumulators to
  // break the min dependency chain; clamp-to-zero hoisted out of the loop
  // (min_j max(d,0) == max(min_j d, 0)).
  v8f rmin0, rmin1;
#pragma unroll
  for (int v = 0; v < 8; ++v) { rmin0[v] = 3.402823466e+38f; rmin1[v] = 3.402823466e+38f; }

  const v2f* __restrict__ ldsBv = (const v2f*)ldsB;
  for (int nt = 0; nt < ntiles; nt += 2) {
    const v2f b0 = ldsBv[nt * 32 + lane];
    const v2f b1 = ldsBv[(nt + 1) * 32 + lane];
    const v8f d0 = __builtin_amdgcn_wmma_f32_16x16x4_f32(
        false, a, false, b0, (short)0, cbase, false, false);
    const v8f d1 = __builtin_amdgcn_wmma_f32_16x16x4_f32(
        false, a, false, b1, (short)0, cbase, false, false);
#pragma unroll
    for (int v = 0; v < 8; ++v) {
      rmin0[v] = fminf(rmin0[v], d0[v]);
      rmin1[v] = fminf(rmin1[v], d1[v]);
    }
  }

  // ---- Reduce: merge accumulators, clamp, then min across each 16-lane half
  // (columns of a row live across one half).
#pragma unroll
  for (int v = 0; v < 8; ++v) {
    float m = fmaxf(fminf(rmin0[v], rmin1[v]), 0.0f);
    m = fminf(m, __shfl_xor(m, 1, 32));
    m = fminf(m, __shfl_xor(m, 2, 32));
    m = fminf(m, __shfl_xor(m, 4, 32));
    m = fminf(m, __shfl_xor(m, 8, 32));
    rmin0[v] = m;
  }

  // Lane 0 holds mins of rows v (0..7); lane 16 holds mins of rows v+8.
  float* __restrict__ ob = outmin + (size_t)b * Nq + mtile * TILE;
  if (lane == 0) {
#pragma unroll
    for (int v = 0; v < 8; ++v) ob[v] = rmin0[v];
  } else if (lane == 16) {
#pragma unroll
    for (int v = 0; v < 8; ++v) ob[v + 8] = rmin0[v];
  }
}

// loss = mean(cham_x) + mean(cham_y); single block, deterministic tree reduce.
__global__ void chamfer_reduce_kernel(const float* __restrict__ cx,
                                      const float* __restrict__ cy,
                                      float* __restrict__ out,
                                      int nx, int ny) {
  __shared__ float s[256];
  const int tid = threadIdx.x;
  float accx = 0.0f, accy = 0.0f;
  for (int i = tid; i < nx; i += 256) accx += cx[i];
  for (int i = tid; i < ny; i += 256) accy += cy[i];
  s[tid] = accx / (float)nx + accy / (float)ny;
  __syncthreads();
  for (int off = 128; off > 0; off >>= 1) {
    if (tid < off) s[tid] += s[tid + off];
    __syncthreads();
  }
  if (tid == 0) out[0] = s[0];
}

extern "C" void kernel_launch(void* const* d_in, const int* in_sizes, int n_in,
                              void* d_out, int out_size, void* d_ws, size_t ws_size,
                              hipStream_t stream) {
  const float* pred = (const float*)d_in[0];  // [B, N1, 3]
  const float* gt   = (const float*)d_in[1];  // [B, N2, 3]
  const int B = 8, N1 = 4096, N2 = 4096;

  float* chamx = (float*)d_ws;                // [B, N1]
  float* chamy = chamx + (size_t)B * N1;      // [B, N2]

  const dim3 block(32 * WAVES_PER_BLOCK);

  // Pass 1: for each pred point, min over gt points -> cham_x
  {
    const dim3 grid(B * (N1 / TILE) / WAVES_PER_BLOCK);
    const size_t shmem = (size_t)(N2 / TILE) * 32 * 2 * sizeof(float);  // 64 KB
    chamfer_min_kernel<<<grid, block, shmem, stream>>>(pred, gt, chamx, N1, N2);
  }
  // Pass 2 (symmetric, pred/gt swapped): min over pred points -> cham_y
  {
    const dim3 grid(B * (N2 / TILE) / WAVES_PER_BLOCK);
    const size_t shmem = (size_t)(N1 / TILE) * 32 * 2 * sizeof(float);  // 64 KB
    chamfer_min_kernel<<<grid, block, shmem, stream>>>(gt, pred, chamy, N2, N1);
  }
  chamfer_reduce_kernel<<<1, 256, 0, stream>>>(chamx, chamy, (float*)d_out,
                                               B * N1, B * N2);
}